// Net_6330781794839
// MI455X (gfx1250) — compile-verified
//
#include <hip/hip_runtime.h>
#include <hip/hip_bf16.h>
#include <math.h>

// ---------------------------------------------------------------------------
// SplineCNN (grid graphs) for MI455X / gfx1250, wave32, WMMA f32 16x16x4.
// Net: spline_conv(1->32) -> pool 6x6 -> spline_conv(32->64) -> pool 4x4
//      -> spline_conv(64->64) -> pool 2x2 -> fc 256->128 -> fc 128->10 -> lsm
// Each spline layer is re-cast as: gather basis-weighted features into
// G[n, 25*f + f] (root appended), then ONE dense GEMM  out = G @ [W;root] + b
// so the heavy math runs on v_wmma_f32_16x16x4_f32.
// ---------------------------------------------------------------------------

#define KCDIV(a, b) (((a) + (b) - 1) / (b))

typedef __attribute__((ext_vector_type(2))) float v2f;
typedef __attribute__((ext_vector_type(8))) float v8f;

__device__ __forceinline__ int imin(int a, int b) { return a < b ? a : b; }
__device__ __forceinline__ int imax(int a, int b) { return a > b ? a : b; }

__device__ __forceinline__ void atomicMaxF(float* addr, float val) {
  unsigned int* a = (unsigned int*)addr;
  unsigned int old = *a, assumed;
  do {
    if (__uint_as_float(old) >= val) break;
    assumed = old;
    old = atomicCAS(a, assumed, __float_as_uint(val));
  } while (old != assumed);
}

__device__ __forceinline__ float eluf(float v) {
  return v > 0.0f ? v : expm1f(v);
}

// ---------------- generic fill ----------------
__global__ void k_fill(float* __restrict__ p, float v, int n) {
  int i = blockIdx.x * blockDim.x + threadIdx.x;
  if (i < n) p[i] = v;
}

// ---------------- pack [Wk ; root ; zero-pad] into Wflat (rows x N) --------
__global__ void k_pack_w(const float* __restrict__ Wk, const float* __restrict__ root,
                         float* __restrict__ Wf, int kfRows, int rootRows,
                         int totRows, int N) {
  int t = blockIdx.x * blockDim.x + threadIdx.x;
  if (t >= totRows * N) return;
  int r = t / N, c = t % N;
  float v = 0.0f;
  if (r < kfRows)                 v = Wk[t];                       // same row-major layout
  else if (r < kfRows + rootRows) v = root[(r - kfRows) * N + c];
  Wf[t] = v;
}

// ---------------- layer-1 edge scatter (f = 1) -----------------------------
__global__ void k_scatter_l1(const float* __restrict__ x, const int* __restrict__ ei,
                             const float* __restrict__ pseudo, float* __restrict__ G,
                             float* __restrict__ cnt, int E) {
  int e = blockIdx.x * blockDim.x + threadIdx.x;
  if (e >= E) return;
  int src = ei[e];
  int dst = ei[E + e];
  float vx = fminf(fmaxf(pseudo[2 * e + 0], 0.0f), 1.0f) * 4.0f;
  float vy = fminf(fmaxf(pseudo[2 * e + 1], 0.0f), 1.0f) * 4.0f;
  int i0x = imin(imax((int)floorf(vx), 0), 4);
  int i0y = imin(imax((int)floorf(vy), 0), 4);
  float fx = vx - (float)i0x, fy = vy - (float)i0y;
  int i1x = imin(i0x + 1, 4), i1y = imin(i0y + 1, 4);
  float val = x[src];
  float* g = G + (size_t)dst * 32;
  atomicAdd(&g[i0x * 5 + i0y], (1.0f - fx) * (1.0f - fy) * val);
  atomicAdd(&g[i0x * 5 + i1y], (1.0f - fx) * fy * val);
  atomicAdd(&g[i1x * 5 + i0y], fx * (1.0f - fy) * val);
  atomicAdd(&g[i1x * 5 + i1y], fx * fy * val);
  atomicAdd(&cnt[dst], 1.0f);
}

// divide aggregated columns by degree; stash x into the root slot (col 25)
__global__ void k_l1_norm(float* __restrict__ G, const float* __restrict__ x,
                          const float* __restrict__ cnt, int n) {
  int i = blockIdx.x * blockDim.x + threadIdx.x;
  if (i >= n) return;
  float invc = 1.0f / fmaxf(cnt[i], 1.0f);
  float* g = G + (size_t)i * 32;
#pragma unroll
  for (int k = 0; k < 25; ++k) g[k] *= invc;
  g[25] = x[i];
}

// ---------------- voxel pool: scatter max + count + position sum -----------
__global__ void k_pool_scatter(const float* __restrict__ hin, const float* __restrict__ pos,
                               float* __restrict__ hp, float* __restrict__ possum,
                               float* __restrict__ cnt, float size, int gy, int gx,
                               int npi, int f, int nNodes) {
  int t = blockIdx.x * blockDim.x + threadIdx.x;
  int node = t / f, ff = t % f;
  if (node >= nNodes) return;
  int b = node / npi;
  float px = pos[2 * node + 0], py = pos[2 * node + 1];
  int cx = imin(imax((int)floorf(px / size), 0), gx - 1);
  int cy = imin(imax((int)floorf(py / size), 0), gy - 1);
  int cell = b * (gy * gx) + cy * gx + cx;
  atomicMaxF(&hp[(size_t)cell * f + ff], hin[(size_t)node * f + ff]);
  if (ff == 0) {
    atomicAdd(&cnt[cell], 1.0f);
    if (possum) {
      atomicAdd(&possum[2 * cell + 0], px);
      atomicAdd(&possum[2 * cell + 1], py);
    }
  }
}

__global__ void k_pool_pos_fin(float* __restrict__ possum, const float* __restrict__ cnt,
                               int ncell) {
  int c = blockIdx.x * blockDim.x + threadIdx.x;
  if (c >= ncell) return;
  float d = fmaxf(cnt[c], 1.0f);
  possum[2 * c + 0] /= d;
  possum[2 * c + 1] /= d;
}

// ---------------- max |cart| over the implicit 8-neighbor grid edges -------
__global__ void k_edge_maxabs(const float* __restrict__ pos, int gh, int gw,
                              float* __restrict__ mx, int ncell) {
  int n = blockIdx.x * blockDim.x + threadIdx.x;
  if (n >= ncell) return;
  const int DY[8] = {-1, -1, -1, 0, 0, 1, 1, 1};
  const int DX[8] = {-1, 0, 1, -1, 1, -1, 0, 1};
  int cells = gh * gw;
  int b = n / cells, r = n % cells, cy = r / gw, cx = r % gw;
  float m = 0.0f;
  float myx = pos[2 * n + 0], myy = pos[2 * n + 1];
#pragma unroll
  for (int d = 0; d < 8; ++d) {
    int sy = cy - DY[d], sx = cx - DX[d];
    if (sy < 0 || sy >= gh || sx < 0 || sx >= gw) continue;
    int s = b * cells + sy * gw + sx;
    m = fmaxf(m, fmaxf(fabsf(myx - pos[2 * s + 0]), fabsf(myy - pos[2 * s + 1])));
  }
  atomicMaxF(mx, m);
}

// ---------------- gather for layers 2/3 (implicit grid edges) --------------
__global__ void k_gather_grid(const float* __restrict__ hin, const float* __restrict__ pos,
                              const float* __restrict__ mx, float* __restrict__ G,
                              int gh, int gw, int f, int ldg, int ncell) {
  int t = blockIdx.x * blockDim.x + threadIdx.x;
  int n = t / f, ff = t % f;
  if (n >= ncell) return;
  const int DY[8] = {-1, -1, -1, 0, 0, 1, 1, 1};
  const int DX[8] = {-1, 0, 1, -1, 1, -1, 0, 1};
  int cells = gh * gw;
  int b = n / cells, r = n % cells, cy = r / gw, cx = r % gw;
  float inv2m = 0.5f / fmaxf(mx[0], 1e-20f);
  int cnt = 0;
#pragma unroll
  for (int d = 0; d < 8; ++d) {
    int sy = cy - DY[d], sx = cx - DX[d];
    if (sy >= 0 && sy < gh && sx >= 0 && sx < gw) ++cnt;
  }
  float invc = 1.0f / (float)imax(cnt, 1);
  float myx = pos[2 * n + 0], myy = pos[2 * n + 1];
  float* g = G + (size_t)n * ldg;
#pragma unroll
  for (int d = 0; d < 8; ++d) {
    int sy = cy - DY[d], sx = cx - DX[d];
    if (sy < 0 || sy >= gh || sx < 0 || sx >= gw) continue;
    int s = b * cells + sy * gw + sx;
    float vx = fminf(fmaxf((myx - pos[2 * s + 0]) * inv2m + 0.5f, 0.0f), 1.0f) * 4.0f;
    float vy = fminf(fmaxf((myy - pos[2 * s + 1]) * inv2m + 0.5f, 0.0f), 1.0f) * 4.0f;
    int i0x = imin(imax((int)floorf(vx), 0), 4);
    int i0y = imin(imax((int)floorf(vy), 0), 4);
    float fx = vx - (float)i0x, fy = vy - (float)i0y;
    int i1x = imin(i0x + 1, 4), i1y = imin(i0y + 1, 4);
    float val = hin[(size_t)s * f + ff] * invc;
    g[(i0x * 5 + i0y) * f + ff] += (1.0f - fx) * (1.0f - fy) * val;
    g[(i0x * 5 + i1y) * f + ff] += (1.0f - fx) * fy * val;
    g[(i1x * 5 + i0y) * f + ff] += fx * (1.0f - fy) * val;
    g[(i1x * 5 + i1y) * f + ff] += fx * fy * val;
  }
  g[25 * f + ff] = hin[(size_t)n * f + ff];  // root-weight slot
}

// ---------------- WMMA GEMM: Out[M,N] = elu(A[M,K] @ B[K,N] + bias) --------
// One wave per 16x16 tile; V_WMMA_F32_16X16X4_F32, f32 accumulate.
// Requires: M%16==0, N%16==0, K%4==0. EXEC all-ones per wave (whole waves
// retire on the tail check, never partial lanes).
__global__ void k_gemm_wmma(const float* __restrict__ A, const float* __restrict__ Bw,
                            const float* __restrict__ bias, float* __restrict__ Out,
                            int M, int Kd, int N, int applyElu) {
  int wave = threadIdx.x >> 5;
  int lane = threadIdx.x & 31;
  int half = lane >> 4;      // 0: K pair {0,1}, 1: K pair {2,3}
  int l16  = lane & 15;
  int tilesN = N >> 4;
  int totalTiles = (M >> 4) * tilesN;
  int tile = blockIdx.x * (blockDim.x >> 5) + wave;
  if (tile >= totalTiles) return;
  int tm = tile / tilesN, tn = tile % tilesN;
  int row0 = tm << 4, col0 = tn << 4;

  const float* arow = A + (size_t)(row0 + l16) * Kd + 2 * half;   // A: 16x4, M=l16
  const float* bcol = Bw + (size_t)(2 * half) * N + col0 + l16;   // B: 4x16, N=l16

  v8f acc = {};
#pragma unroll 4
  for (int k = 0; k < Kd; k += 4) {
    v2f a, b;
    a.x = arow[k];
    a.y = arow[k + 1];
    b.x = bcol[(size_t)k * N];
    b.y = bcol[(size_t)(k + 1) * N];
    acc = __builtin_amdgcn_wmma_f32_16x16x4_f32(false, a, false, b,
                                                (short)0, acc, false, false);
  }
  float bb = bias ? bias[col0 + l16] : 0.0f;
#pragma unroll
  for (int i = 0; i < 8; ++i) {
    float v = acc[i] + bb;
    if (applyElu) v = eluf(v);
    Out[(size_t)(row0 + i + 8 * half) * N + col0 + l16] = v;   // C/D: M=i+8*half
  }
}

// ---------------- fc2 (128->10) + log_softmax ------------------------------
__global__ void k_fc2_lsm(const float* __restrict__ hin, const float* __restrict__ w,
                          const float* __restrict__ b, float* __restrict__ out) {
  int row = blockIdx.x * blockDim.x + threadIdx.x;
  if (row >= 256) return;
  float o[10];
#pragma unroll
  for (int j = 0; j < 10; ++j) o[j] = b[j];
  for (int i = 0; i < 128; ++i) {
    float h = hin[row * 128 + i];
#pragma unroll
    for (int j = 0; j < 10; ++j) o[j] += h * w[i * 10 + j];
  }
  float m = o[0];
#pragma unroll
  for (int j = 1; j < 10; ++j) m = fmaxf(m, o[j]);
  float s = 0.0f;
#pragma unroll
  for (int j = 0; j < 10; ++j) s += expf(o[j] - m);
  float ls = logf(s) + m;
#pragma unroll
  for (int j = 0; j < 10; ++j) out[row * 10 + j] = o[j] - ls;
}

// ---------------------------------------------------------------------------
extern "C" void kernel_launch(void* const* d_in, const int* in_sizes, int n_in,
                              void* d_out, int out_size, void* d_ws, size_t ws_size,
                              hipStream_t stream) {
  const float* x      = (const float*)d_in[0];
  const float* pos    = (const float*)d_in[1];
  const float* pseudo = (const float*)d_in[2];
  const float* W1     = (const float*)d_in[3];
  const float* root1  = (const float*)d_in[4];
  const float* b1     = (const float*)d_in[5];
  const float* W2     = (const float*)d_in[6];
  const float* root2  = (const float*)d_in[7];
  const float* b2     = (const float*)d_in[8];
  const float* W3     = (const float*)d_in[9];
  const float* root3  = (const float*)d_in[10];
  const float* b3     = (const float*)d_in[11];
  const float* fc1w   = (const float*)d_in[12];
  const float* fc1b   = (const float*)d_in[13];
  const float* fc2w   = (const float*)d_in[14];
  const float* fc2b   = (const float*)d_in[15];
  const int*   ei     = (const int*)d_in[16];
  const int E = in_sizes[16] / 2;

  const int N1  = 256 * 28 * 28;  // 200704 nodes, layer 1
  const int NC1 = 256 * 36;       // 9216 cells after pool1 (6x6)
  const int NC2 = 256 * 16;       // 4096 cells after pool2 (4x4)
  const int NC3 = 256 * 4;        // 1024 cells after pool3 (2x2)
  const int LD1 = 32;             // 25 + root(1), padded to 32
  const int LD2 = 25 * 32 + 32;   // 832
  const int LD3 = 25 * 64 + 64;   // 1664

  char* ws = (char*)d_ws;
  size_t off = 0;
  auto alloc = [&](size_t bytes) -> float* {
    float* p = (float*)(ws + off);
    off += (bytes + 255) & ~(size_t)255;
    return p;
  };

  float* G1    = alloc((size_t)N1 * LD1 * 4);
  float* cnt1  = alloc((size_t)N1 * 4);
  float* Wf1   = alloc((size_t)32 * 32 * 4);
  float* h1    = alloc((size_t)N1 * 32 * 4);
  float* h1p   = alloc((size_t)NC1 * 32 * 4);
  float* pos1  = alloc((size_t)NC1 * 2 * 4);
  float* cp1   = alloc((size_t)NC1 * 4);
  float* mx1   = alloc(16);
  float* G2    = alloc((size_t)NC1 * LD2 * 4);
  float* Wf2   = alloc((size_t)LD2 * 64 * 4);
  float* h2    = alloc((size_t)NC1 * 64 * 4);
  float* h2p   = alloc((size_t)NC2 * 64 * 4);
  float* pos2  = alloc((size_t)NC2 * 2 * 4);
  float* cp2   = alloc((size_t)NC2 * 4);
  float* mx2   = alloc(16);
  float* G3    = alloc((size_t)NC2 * LD3 * 4);
  float* Wf3   = alloc((size_t)LD3 * 64 * 4);
  float* h3    = alloc((size_t)NC2 * 64 * 4);
  float* h3p   = alloc((size_t)NC3 * 64 * 4);   // == (256 x 256) row-major
  float* fc1o  = alloc((size_t)256 * 128 * 4);
  float* cp3   = alloc((size_t)NC3 * 4);
  (void)ws_size; (void)n_in; (void)out_size;

  const int T = 256;
  const float NEG = -3.0e38f;

  // ---------------- Layer 1: scatter -> G1, GEMM 200704 x 32 x 32 ----------
  k_fill<<<KCDIV(N1 * LD1, T), T, 0, stream>>>(G1, 0.0f, N1 * LD1);
  k_fill<<<KCDIV(N1, T), T, 0, stream>>>(cnt1, 0.0f, N1);
  k_pack_w<<<KCDIV(32 * 32, T), T, 0, stream>>>(W1, root1, Wf1, 25, 1, 32, 32);
  k_scatter_l1<<<KCDIV(E, T), T, 0, stream>>>(x, ei, pseudo, G1, cnt1, E);
  k_l1_norm<<<KCDIV(N1, T), T, 0, stream>>>(G1, x, cnt1, N1);
  k_gemm_wmma<<<KCDIV((N1 / 16) * (32 / 16), 8), T, 0, stream>>>(
      G1, Wf1, b1, h1, N1, LD1, 32, 1);

  // ---------------- Pool 1: 28x28 -> 6x6 ----------------------------------
  k_fill<<<KCDIV(NC1 * 32, T), T, 0, stream>>>(h1p, NEG, NC1 * 32);
  k_fill<<<KCDIV(NC1 * 2, T), T, 0, stream>>>(pos1, 0.0f, NC1 * 2);
  k_fill<<<KCDIV(NC1, T), T, 0, stream>>>(cp1, 0.0f, NC1);
  k_pool_scatter<<<KCDIV(N1 * 32, T), T, 0, stream>>>(
      h1, pos, h1p, pos1, cp1, 5.0f, 6, 6, 784, 32, N1);
  k_pool_pos_fin<<<KCDIV(NC1, T), T, 0, stream>>>(pos1, cp1, NC1);

  // ---------------- Layer 2: gather -> G2, GEMM 9216 x 832 x 64 ------------
  k_fill<<<1, 32, 0, stream>>>(mx1, 0.0f, 1);
  k_edge_maxabs<<<KCDIV(NC1, T), T, 0, stream>>>(pos1, 6, 6, mx1, NC1);
  k_fill<<<KCDIV(NC1 * LD2, T), T, 0, stream>>>(G2, 0.0f, NC1 * LD2);
  k_gather_grid<<<KCDIV(NC1 * 32, T), T, 0, stream>>>(
      h1p, pos1, mx1, G2, 6, 6, 32, LD2, NC1);
  k_pack_w<<<KCDIV(LD2 * 64, T), T, 0, stream>>>(W2, root2, Wf2, 800, 32, LD2, 64);
  k_gemm_wmma<<<KCDIV((NC1 / 16) * (64 / 16), 8), T, 0, stream>>>(
      G2, Wf2, b2, h2, NC1, LD2, 64, 1);

  // ---------------- Pool 2: 6x6 -> 4x4 ------------------------------------
  k_fill<<<KCDIV(NC2 * 64, T), T, 0, stream>>>(h2p, NEG, NC2 * 64);
  k_fill<<<KCDIV(NC2 * 2, T), T, 0, stream>>>(pos2, 0.0f, NC2 * 2);
  k_fill<<<KCDIV(NC2, T), T, 0, stream>>>(cp2, 0.0f, NC2);
  k_pool_scatter<<<KCDIV(NC1 * 64, T), T, 0, stream>>>(
      h2, pos1, h2p, pos2, cp2, 7.0f, 4, 4, 36, 64, NC1);
  k_pool_pos_fin<<<KCDIV(NC2, T), T, 0, stream>>>(pos2, cp2, NC2);

  // ---------------- Layer 3: gather -> G3, GEMM 4096 x 1664 x 64 -----------
  k_fill<<<1, 32, 0, stream>>>(mx2, 0.0f, 1);
  k_edge_maxabs<<<KCDIV(NC2, T), T, 0, stream>>>(pos2, 4, 4, mx2, NC2);
  k_fill<<<KCDIV(NC2 * LD3, T), T, 0, stream>>>(G3, 0.0f, NC2 * LD3);
  k_gather_grid<<<KCDIV(NC2 * 64, T), T, 0, stream>>>(
      h2p, pos2, mx2, G3, 4, 4, 64, LD3, NC2);
  k_pack_w<<<KCDIV(LD3 * 64, T), T, 0, stream>>>(W3, root3, Wf3, 1600, 64, LD3, 64);
  k_gemm_wmma<<<KCDIV((NC2 / 16) * (64 / 16), 8), T, 0, stream>>>(
      G3, Wf3, b3, h3, NC2, LD3, 64, 1);

  // ---------------- Pool 3: 4x4 -> 2x2 (features only) ---------------------
  k_fill<<<KCDIV(NC3 * 64, T), T, 0, stream>>>(h3p, NEG, NC3 * 64);
  k_fill<<<KCDIV(NC3, T), T, 0, stream>>>(cp3, 0.0f, NC3);
  k_pool_scatter<<<KCDIV(NC2 * 64, T), T, 0, stream>>>(
      h3, pos2, h3p, nullptr, cp3, 14.0f, 2, 2, 16, 64, NC2);

  // ---------------- FC head ------------------------------------------------
  // h3p is exactly (256 x 256) row-major: cells in cid order, 64 feats each.
  k_gemm_wmma<<<KCDIV((256 / 16) * (128 / 16), 8), T, 0, stream>>>(
      h3p, fc1w, fc1b, fc1o, 256, 256, 128, 1);
  k_fc2_lsm<<<1, 256, 0, stream>>>(fc1o, fc2w, fc2b, (float*)d_out);
}